// NPLinear_52055003627834
// MI455X (gfx1250) — compile-verified
//
#include <hip/hip_runtime.h>

typedef float fv2 __attribute__((ext_vector_type(2)));
typedef float fv8 __attribute__((ext_vector_type(8)));

// ---------------------------------------------------------------------------
// Pass A: per-layer means.  grid = (32 bc, 8 slabs), block = 256 (8 waves).
// Each block handles rows [slab*H/8, (slab+1)*H/8) of one (b,c) matrix.
// Produces col_means directly, partial row sums + partial rc sums to scratch.
// ---------------------------------------------------------------------------
template <int W>
__global__ __launch_bounds__(256) void means_kernel(
    const float* __restrict__ wt, float* __restrict__ rowpart,
    float* __restrict__ col_out, float* __restrict__ rcpart, int H) {
  constexpr int ITERS = W / 128;
  __shared__ float sm[8 * W];
  __shared__ float src_[8];
  const int bc = blockIdx.x;
  const int slab = blockIdx.y;
  const int wid = threadIdx.x >> 5;
  const int lane = threadIdx.x & 31;
  const int wids = __builtin_amdgcn_readfirstlane(wid);  // scalar wave id
  const int hs = H >> 3;
  const int h0 = slab * hs;
  const float* m = wt + (size_t)bc * (size_t)H * W;
  const unsigned lo = (unsigned)lane * 4;  // per-lane float offset

  float racc[ITERS][4];
#pragma unroll
  for (int i = 0; i < ITERS; ++i)
#pragma unroll
    for (int q = 0; q < 4; ++q) racc[i][q] = 0.f;
  float rcacc = 0.f;

#pragma unroll 1
  for (int h = h0 + wids; h < h0 + hs; h += 8) {  // scalar loop
    const float4* rowp =
        reinterpret_cast<const float4*>(m + (size_t)h * W + lo);
    float rsum = 0.f;
#pragma unroll
    for (int it = 0; it < ITERS; ++it) {
      float4 v = rowp[it * 32];  // imm offset it*512B
      racc[it][0] += v.x; racc[it][1] += v.y;
      racc[it][2] += v.z; racc[it][3] += v.w;
      rsum += (v.x + v.y) + (v.z + v.w);
    }
#pragma unroll
    for (int d = 16; d >= 1; d >>= 1) rsum += __shfl_xor(rsum, d, 32);
    if (lane == 0) {
      col_out[(size_t)bc * H + h] = rsum * (1.0f / (float)W);
      rcacc += rsum;
    }
  }
#pragma unroll
  for (int it = 0; it < ITERS; ++it) {
    float4 v = make_float4(racc[it][0], racc[it][1], racc[it][2], racc[it][3]);
    *reinterpret_cast<float4*>(&sm[wids * W + lo + it * 128]) = v;
  }
  if (lane == 0) src_[wids] = rcacc;
  __syncthreads();
  for (int j = threadIdx.x; j < W; j += 256) {
    float s = 0.f;
#pragma unroll
    for (int v = 0; v < 8; ++v) s += sm[v * W + j];
    rowpart[((size_t)bc * 8 + slab) * W + j] = s;
  }
  if (threadIdx.x == 0) {
    float s = 0.f;
#pragma unroll
    for (int v = 0; v < 8; ++v) s += src_[v];
    rcpart[bc * 8 + slab] = s;
  }
}

// Finalize row means + rc means across the 8 slabs.
__global__ void rowfin_kernel(const float* __restrict__ rowpart,
                              const float* __restrict__ rcpart,
                              float* __restrict__ row_out,
                              float* __restrict__ rc_out, int W, int H) {
  int idx = blockIdx.x * blockDim.x + threadIdx.x;
  if (idx >= 32 * W) return;
  int bc = idx / W, j = idx % W;
  float s = 0.f;
#pragma unroll
  for (int t = 0; t < 8; ++t) s += rowpart[((size_t)bc * 8 + t) * W + j];
  row_out[(size_t)bc * W + j] = s * (1.0f / (float)H);
  if (j == 0) {
    float r = 0.f;
#pragma unroll
    for (int t = 0; t < 8; ++t) r += rcpart[bc * 8 + t];
    rc_out[bc] = r * (1.0f / ((float)H * (float)W));
  }
}

// Bias means: one wave per (layer, b, c): 96 waves.
__global__ void bmeans_kernel(const float* __restrict__ b0,
                              const float* __restrict__ b1,
                              const float* __restrict__ b2, int n0, int n1,
                              int n2, float* __restrict__ bm) {
  int g = blockIdx.x * (blockDim.x >> 5) + (threadIdx.x >> 5);
  int lane = threadIdx.x & 31;
  if (g >= 96) return;
  int i = g >> 5, bc = g & 31;
  const float* p = (i == 0) ? b0 : (i == 1) ? b1 : b2;
  int n = (i == 0) ? n0 : (i == 1) ? n1 : n2;
  float s = 0.f;
  for (int j = lane; j < n; j += 32) s += p[(size_t)bc * n + j];
#pragma unroll
  for (int d = 16; d >= 1; d >>= 1) s += __shfl_xor(s, d, 32);
  if (lane == 0) bm[g] = s * (1.0f / (float)n);
}

struct SArgs {
  const float* rcw[3];
  const float* rcb[3];
  const float* Lb[3];
  const float* brcw[3];
  const float* brcb[3];
};

// Per (layer,b,o): zc = Lb + z2 ;  u2 = wb @ brcw^T + brcb.  (96 threads)
__global__ void scalars_kernel(SArgs sa, const float* __restrict__ rc,
                               const float* __restrict__ bm,
                               float* __restrict__ zc, float* __restrict__ u2) {
  int t = threadIdx.x;
  if (t >= 96) return;
  int i = t >> 5, rem = t & 31, b = rem >> 4, o = rem & 15;
  float z2 = 0.f, uu = 0.f;
  for (int k = 0; k < 96; ++k) {
    int grp = k >> 4, cc = k & 15;
    float wbk = (k < 48) ? rc[grp * 32 + b * 16 + cc]
                         : bm[(grp - 3) * 32 + b * 16 + cc];
    z2 += sa.rcw[i][o * 96 + k] * wbk;
    uu += sa.brcw[i][o * 96 + k] * wbk;
  }
  zc[t] = z2 + sa.rcb[i][o] + sa.Lb[i][o];
  u2[t] = uu + sa.brcb[i][o];
}

// Per layer: C3[b,o,n_in] = zc + rb + rw@rcat ; C4[b,o,n_out] = cb + cw@ccat ;
// out_b = bb + u2 + bw@ccat.   One thread per (b, n).
__global__ void cvec_kernel(
    const float* __restrict__ rw, const float* __restrict__ rb,
    const float* __restrict__ cw, const float* __restrict__ cb,
    const float* __restrict__ bw, const float* __restrict__ bb,
    const float* __restrict__ zc, const float* __restrict__ u2,
    const float* __restrict__ row_this, const float* __restrict__ row_next,
    const float* __restrict__ col_this, const float* __restrict__ col_prev,
    const float* __restrict__ bias_this, const float* __restrict__ bias_prev,
    int n_i, int n_o, int row_in, int col_in, float* __restrict__ C3,
    float* __restrict__ C4, float* __restrict__ outb) {
  int idx = blockIdx.x * blockDim.x + threadIdx.x;
  int nmax = n_i > n_o ? n_i : n_o;
  if (idx >= 2 * nmax) return;
  int b = idx / nmax, n = idx % nmax;
  if (n < n_i) {
    float vals[48];
    for (int c = 0; c < 16; ++c) vals[c] = row_this[(size_t)(b * 16 + c) * n_i + n];
    if (row_in > 16) {
      for (int c = 0; c < 16; ++c) {
        vals[16 + c] = col_prev[(size_t)(b * 16 + c) * n_i + n];
        vals[32 + c] = bias_prev[(size_t)(b * 16 + c) * n_i + n];
      }
    }
    for (int o = 0; o < 16; ++o) {
      float acc = zc[b * 16 + o] + rb[o];
      for (int c = 0; c < row_in; ++c) acc += rw[o * row_in + c] * vals[c];
      C3[(size_t)(b * 16 + o) * n_i + n] = acc;
    }
  }
  if (n < n_o) {
    float vals[48];
    for (int c = 0; c < 16; ++c) {
      vals[c] = col_this[(size_t)(b * 16 + c) * n_o + n];
      vals[16 + c] = bias_this[(size_t)(b * 16 + c) * n_o + n];
    }
    if (col_in > 32)
      for (int c = 0; c < 16; ++c)
        vals[32 + c] = row_next[(size_t)(b * 16 + c) * n_o + n];
    for (int o = 0; o < 16; ++o) {
      float a4 = cb[o];
      float ab = bb[o] + u2[b * 16 + o];
      for (int c = 0; c < col_in; ++c) {
        a4 += cw[o * col_in + c] * vals[c];
        ab += bw[o * col_in + c] * vals[c];
      }
      C4[(size_t)(b * 16 + o) * n_o + n] = a4;
      outb[(size_t)(b * 16 + o) * n_o + n] = ab;
    }
  }
}

// ---------------------------------------------------------------------------
// Pass C (dominant, memory bound): out[b,o,h,p] =
//   WMMA16x16(K=16 via 4x V_WMMA_F32_16X16X4_F32)(Lw, w[b,:,h,p-tile])
//   with C operand preloaded as C3[b,o,p] + C4[b,o,h].
// Strategy: 24 persistent per-lane pointer pairs set up once per chunk; the
// fully-unrolled 2-tile loop uses only constant element offsets, so every
// access is `global_{load,store}_b32 v, v[pair], off offset:imm` with zero
// per-tile address math.  CHUNK=2 keeps the live set well under 256 VGPRs.
// A layout (16x4 f32): lanes 0-15 M=o,K={0,1}; lanes 16-31 M=o,K={2,3}.
// B layout (4x16 f32): lanes 0-15 N=px,K={0,1}; lanes 16-31 N=px,K={2,3}.
// D layout: VGPR r -> M=r (lanes 0-15) / M=r+8 (lanes 16-31), N=lane&15.
// ---------------------------------------------------------------------------
__global__ __launch_bounds__(256) void mix_kernel(
    const float* __restrict__ w, const float* __restrict__ Lw,
    const float* __restrict__ C3, const float* __restrict__ C4,
    float* __restrict__ out, int H, int W) {
  const int lane = threadIdx.x & 31;
  const int wid = threadIdx.x >> 5;
  const int half = lane >> 4;
  const unsigned l16 = (unsigned)(lane & 15);
  const size_t HW = (size_t)H * W;
  const unsigned HWu = (unsigned)HW;

  // A fragments: all of Lw (16x16), split into 4 K-blocks of 4.
  fv2 a[4];
#pragma unroll
  for (int kb = 0; kb < 4; ++kb) {
    int c0 = kb * 4 + half * 2;
    a[kb][0] = Lw[(lane & 15) * 16 + c0];
    a[kb][1] = Lw[(lane & 15) * 16 + c0 + 1];
  }

  // Per-lane 32-bit element offsets (loop-invariant).
  unsigned boff[8], soff[8], c3off[8], c4off[8];
#pragma unroll
  for (int kb = 0; kb < 4; ++kb) {
    unsigned c0 = (unsigned)(kb * 4 + half * 2);
    boff[2 * kb] = c0 * HWu + l16;
    boff[2 * kb + 1] = boff[2 * kb] + HWu;
  }
#pragma unroll
  for (int r = 0; r < 8; ++r) {
    unsigned o = (unsigned)(r + 8 * half);
    soff[r] = o * HWu + l16;
    c3off[r] = o * (unsigned)W + l16;
    c4off[r] = o * (unsigned)H;
  }

  const int CHUNK = 2;               // 16-px tiles per chunk (W % 32 == 0)
  const int cpr = (W >> 4) / CHUNK;  // chunks per row
  const int totalChunks = 2 * H * cpr;
  const int gw = __builtin_amdgcn_readfirstlane(blockIdx.x * 8 + wid);
  const int gstride = gridDim.x * 8;

#pragma unroll 1
  for (int ch = gw; ch < totalChunks; ch += gstride) {  // scalar loop
    const int b = ch / (H * cpr);
    const int rem = ch - b * (H * cpr);
    const int h = rem / cpr;
    const int p0 = (rem - h * cpr) * (CHUNK * 16);

    // Scalar (SGPR) bases for this chunk.
    const float* wb_ = w + (size_t)b * 16 * HW + (size_t)h * W + p0;
    float* ob_ = out + (size_t)b * 16 * HW + (size_t)h * W + p0;
    const float* C3b = C3 + (size_t)b * 16 * W + p0;
    const float* C4b = C4 + (size_t)b * 16 * H + h;

    // Persistent per-lane pointers for this chunk (address pairs in VGPRs).
    const float* bp[8];
    const float* cp[8];
    float* sp[8];
#pragma unroll
    for (int k = 0; k < 8; ++k) bp[k] = wb_ + boff[k];
#pragma unroll
    for (int r = 0; r < 8; ++r) {
      cp[r] = C3b + c3off[r];
      sp[r] = ob_ + soff[r];
    }

    // Row addend (z4) per output channel, constant over the chunk.
    float c4v[8];
#pragma unroll
    for (int r = 0; r < 8; ++r) c4v[r] = C4b[c4off[r]];

#pragma unroll
    for (int t = 0; t < CHUNK; ++t) {
      // C operand = C3 + C4 addends.
      fv8 acc;
#pragma unroll
      for (int r = 0; r < 8; ++r) acc[r] = c4v[r] + cp[r][t * 16];
      // B fragments: 16 channels x 16 pixels of w.
      fv2 bf[4];
#pragma unroll
      for (int kb = 0; kb < 4; ++kb) {
        bf[kb][0] = bp[2 * kb][t * 16];
        bf[kb][1] = bp[2 * kb + 1][t * 16];
      }
      acc = __builtin_amdgcn_wmma_f32_16x16x4_f32(false, a[0], false, bf[0],
                                                  (short)0, acc, false, false);
      acc = __builtin_amdgcn_wmma_f32_16x16x4_f32(false, a[1], false, bf[1],
                                                  (short)0, acc, false, false);
      acc = __builtin_amdgcn_wmma_f32_16x16x4_f32(false, a[2], false, bf[2],
                                                  (short)0, acc, false, false);
      acc = __builtin_amdgcn_wmma_f32_16x16x4_f32(false, a[3], false, bf[3],
                                                  (short)0, acc, false, false);
#pragma unroll
      for (int r = 0; r < 8; ++r)
        __builtin_nontemporal_store(acc[r], sp[r] + t * 16);
    }
  }
}

// ---------------------------------------------------------------------------
extern "C" void kernel_launch(void* const* d_in, const int* in_sizes, int n_in,
                              void* d_out, int out_size, void* d_ws,
                              size_t ws_size, hipStream_t stream) {
  (void)in_sizes; (void)n_in; (void)out_size; (void)ws_size;
  const int DIMS[4] = {512, 1024, 1024, 512};
  const int ROW_IN[3] = {16, 48, 48};
  const int COL_IN[3] = {48, 48, 32};

  const float* w[3] = {(const float*)d_in[0], (const float*)d_in[2],
                       (const float*)d_in[4]};
  const float* bias[3] = {(const float*)d_in[1], (const float*)d_in[3],
                          (const float*)d_in[5]};
  const float *Lw[3], *Lb[3], *rcw[3], *rcb[3], *rw[3], *rb[3], *cw[3], *cb[3],
      *bw[3], *bb[3], *brcw[3], *brcb[3];
  for (int i = 0; i < 3; ++i) {
    int base = 6 + 12 * i;
    Lw[i] = (const float*)d_in[base + 0];
    Lb[i] = (const float*)d_in[base + 1];
    rcw[i] = (const float*)d_in[base + 2];
    rcb[i] = (const float*)d_in[base + 3];
    rw[i] = (const float*)d_in[base + 4];
    rb[i] = (const float*)d_in[base + 5];
    cw[i] = (const float*)d_in[base + 6];
    cb[i] = (const float*)d_in[base + 7];
    bw[i] = (const float*)d_in[base + 8];
    bb[i] = (const float*)d_in[base + 9];
    brcw[i] = (const float*)d_in[base + 10];
    brcb[i] = (const float*)d_in[base + 11];
  }

  float* ws = (float*)d_ws;
  size_t off = 0;
  float* rowm[3];
  float* colm[3];
  float* C3[3];
  float* C4[3];
  for (int i = 0; i < 3; ++i) { rowm[i] = ws + off; off += (size_t)32 * DIMS[i]; }
  for (int i = 0; i < 3; ++i) { colm[i] = ws + off; off += (size_t)32 * DIMS[i + 1]; }
  float* rc = ws + off; off += 96;
  float* bm = ws + off; off += 96;
  float* zc = ws + off; off += 96;
  float* u2 = ws + off; off += 96;
  for (int i = 0; i < 3; ++i) { C3[i] = ws + off; off += (size_t)32 * DIMS[i]; }
  for (int i = 0; i < 3; ++i) { C4[i] = ws + off; off += (size_t)32 * DIMS[i + 1]; }
  float* rowpart = ws + off; off += (size_t)32 * 8 * 1024;
  float* rcpart = ws + off; off += 32 * 8;

  float* out = (float*)d_out;
  float* outw[3];
  float* outb[3];
  size_t oo = 0;
  for (int i = 0; i < 3; ++i) {
    outw[i] = out + oo;
    oo += (size_t)32 * DIMS[i + 1] * DIMS[i];
  }
  for (int i = 0; i < 3; ++i) {
    outb[i] = out + oo;
    oo += (size_t)32 * DIMS[i + 1];
  }

  // Pass A: means (one full read of each w).
  for (int i = 0; i < 3; ++i) {
    int H = DIMS[i + 1], W = DIMS[i];
    dim3 g(32, 8);
    if (W == 512)
      means_kernel<512><<<g, 256, 0, stream>>>(w[i], rowpart, colm[i], rcpart, H);
    else
      means_kernel<1024><<<g, 256, 0, stream>>>(w[i], rowpart, colm[i], rcpart, H);
    int tot = 32 * W;
    rowfin_kernel<<<(tot + 255) / 256, 256, 0, stream>>>(rowpart, rcpart,
                                                         rowm[i], rc + i * 32,
                                                         W, H);
  }
  bmeans_kernel<<<12, 256, 0, stream>>>(bias[0], bias[1], bias[2], DIMS[1],
                                        DIMS[2], DIMS[3], bm);
  SArgs sa;
  for (int i = 0; i < 3; ++i) {
    sa.rcw[i] = rcw[i]; sa.rcb[i] = rcb[i]; sa.Lb[i] = Lb[i];
    sa.brcw[i] = brcw[i]; sa.brcb[i] = brcb[i];
  }
  scalars_kernel<<<1, 128, 0, stream>>>(sa, rc, bm, zc, u2);

  for (int i = 0; i < 3; ++i) {
    int n_i = DIMS[i], n_o = DIMS[i + 1];
    int nmax = n_i > n_o ? n_i : n_o;
    cvec_kernel<<<(2 * nmax + 255) / 256, 256, 0, stream>>>(
        rw[i], rb[i], cw[i], cb[i], bw[i], bb[i], zc + i * 32, u2 + i * 32,
        rowm[i], (i < 2) ? rowm[i + 1] : nullptr, colm[i],
        (i > 0) ? colm[i - 1] : nullptr, bias[i],
        (i > 0) ? bias[i - 1] : nullptr, n_i, n_o, ROW_IN[i], COL_IN[i], C3[i],
        C4[i], outb[i]);
  }

  // Pass C in reverse layer order: w2 was read most recently in pass A and is
  // most likely still resident in the 192 MB L2.
  for (int i = 2; i >= 0; --i) {
    mix_kernel<<<1024, 256, 0, stream>>>(w[i], Lw[i], C3[i], C4[i], outw[i],
                                         DIMS[i + 1], DIMS[i]);
  }
}